// VarifoldLoss_77163382440708
// MI455X (gfx1250) — compile-verified
//
#include <hip/hip_runtime.h>
#include <math.h>

// ---------------- problem constants (from reference) ----------------
#define BATCH 4
#define NPTS  4096
#define KNN   10
// sigma = 1.0 -> kernel element = exp(-(d2)) * (n.m)^2

typedef __attribute__((ext_vector_type(2))) float v2f;
typedef __attribute__((ext_vector_type(8))) float v8f;

// =====================================================================
// Kernel 1: kNN (k=10) + covariance + smallest eigenvector per point.
// One thread per query point; candidate points staged through LDS.
// =====================================================================
__global__ __launch_bounds__(256) void estimate_normals_kernel(
    const float* __restrict__ xyz, float* __restrict__ nor)
{
    const int blocks_per_batch = NPTS / 256;
    const int b    = blockIdx.x / blocks_per_batch;
    const int base = (blockIdx.x % blocks_per_batch) * 256;
    const int qi   = base + threadIdx.x;

    const float* cloud = xyz + (size_t)b * NPTS * 3;
    const float qx = cloud[qi * 3 + 0];
    const float qy = cloud[qi * 3 + 1];
    const float qz = cloud[qi * 3 + 2];

    float dk[KNN];
    int   id[KNN];
#pragma unroll
    for (int m = 0; m < KNN; ++m) { dk[m] = 3.4e38f; id[m] = 0; }

    __shared__ float sx[256], sy[256], sz[256];

    for (int tile = 0; tile < NPTS; tile += 256) {
        __syncthreads();
        const int j = tile + threadIdx.x;
        sx[threadIdx.x] = cloud[j * 3 + 0];
        sy[threadIdx.x] = cloud[j * 3 + 1];
        sz[threadIdx.x] = cloud[j * 3 + 2];
        __syncthreads();

        for (int t = 0; t < 256; ++t) {
            const float dx = qx - sx[t];
            const float dy = qy - sy[t];
            const float dz = qz - sz[t];
            const float d2 = dx * dx + dy * dy + dz * dz;
            if (d2 < dk[KNN - 1]) {
                dk[KNN - 1] = d2; id[KNN - 1] = tile + t;
                // one bubble pass restores sorted order (array was sorted)
#pragma unroll
                for (int m = KNN - 1; m > 0; --m) {
                    if (dk[m] < dk[m - 1]) {
                        float td = dk[m]; dk[m] = dk[m - 1]; dk[m - 1] = td;
                        int   ti = id[m]; id[m] = id[m - 1]; id[m - 1] = ti;
                    }
                }
            }
        }
    }

    // gather neighbors, mean, covariance (6 unique entries)
    float px[KNN], py[KNN], pz[KNN];
    float mx = 0.f, my = 0.f, mz = 0.f;
#pragma unroll
    for (int m = 0; m < KNN; ++m) {
        const float* p = cloud + (size_t)id[m] * 3;
        px[m] = p[0]; py[m] = p[1]; pz[m] = p[2];
        mx += px[m]; my += py[m]; mz += pz[m];
    }
    const float invk = 1.0f / (float)KNN;
    mx *= invk; my *= invk; mz *= invk;

    float c00 = 0.f, c01 = 0.f, c02 = 0.f, c11 = 0.f, c12 = 0.f, c22 = 0.f;
#pragma unroll
    for (int m = 0; m < KNN; ++m) {
        const float dx = px[m] - mx, dy = py[m] - my, dz = pz[m] - mz;
        c00 += dx * dx; c01 += dx * dy; c02 += dx * dz;
        c11 += dy * dy; c12 += dy * dz; c22 += dz * dz;
    }
    c00 *= invk; c01 *= invk; c02 *= invk; c11 *= invk; c12 *= invk; c22 *= invk;

    // smallest eigenvalue of symmetric 3x3 (trigonometric method)
    float nx = 0.f, ny = 0.f, nz = 1.f;
    const float q  = (c00 + c11 + c22) * (1.0f / 3.0f);
    const float p1 = c01 * c01 + c02 * c02 + c12 * c12;
    const float a00 = c00 - q, a11 = c11 - q, a22 = c22 - q;
    const float p2 = a00 * a00 + a11 * a11 + a22 * a22 + 2.0f * p1;
    const float p  = sqrtf(p2 * (1.0f / 6.0f));
    if (p > 1e-12f) {
        const float ip = 1.0f / p;
        const float b00 = a00 * ip, b01 = c01 * ip, b02 = c02 * ip;
        const float b11 = a11 * ip, b12 = c12 * ip, b22 = a22 * ip;
        float detB = b00 * (b11 * b22 - b12 * b12)
                   - b01 * (b01 * b22 - b12 * b02)
                   + b02 * (b01 * b12 - b11 * b02);
        float r = fminf(1.0f, fmaxf(-1.0f, 0.5f * detB));
        const float phi = acosf(r) * (1.0f / 3.0f);
        const float lam = q + 2.0f * p * cosf(phi + 2.0943951023931953f); // smallest
        // eigenvector: largest cross product of rows of (C - lam*I)
        const float r0x = c00 - lam, r0y = c01,       r0z = c02;
        const float r1x = c01,       r1y = c11 - lam, r1z = c12;
        const float r2x = c02,       r2y = c12,       r2z = c22 - lam;
        float e0x = r0y * r1z - r0z * r1y, e0y = r0z * r1x - r0x * r1z, e0z = r0x * r1y - r0y * r1x;
        float e1x = r0y * r2z - r0z * r2y, e1y = r0z * r2x - r0x * r2z, e1z = r0x * r2y - r0y * r2x;
        float e2x = r1y * r2z - r1z * r2y, e2y = r1z * r2x - r1x * r2z, e2z = r1x * r2y - r1y * r2x;
        const float n0 = e0x * e0x + e0y * e0y + e0z * e0z;
        const float n1 = e1x * e1x + e1y * e1y + e1z * e1z;
        const float n2 = e2x * e2x + e2y * e2y + e2z * e2z;
        float bx = e0x, by = e0y, bz = e0z, bn = n0;
        if (n1 > bn) { bx = e1x; by = e1y; bz = e1z; bn = n1; }
        if (n2 > bn) { bx = e2x; by = e2y; bz = e2z; bn = n2; }
        if (bn > 1e-20f) {
            const float inv = rsqrtf(bn);
            nx = bx * inv; ny = by * inv; nz = bz * inv;
        }
    }

    float* np = nor + ((size_t)b * NPTS + qi) * 3;
    np[0] = nx; np[1] = ny; np[2] = nz;
}

// =====================================================================
// Kernel 2: varifold pairwise reduction with V_WMMA_F32_16X16X4_F32.
// One wave per (batch, 16-row i-tile); loops over all 256 j-tiles.
//   G1[M,N] = A1 x B1, A1 row = (2x0,2x1,2x2,-|x|^2), B1 col = (y0,y1,y2,1)
//           = 2 x.y - |x|^2        -> exp(G1 - |y|^2) = exp(-d2)
//   G2[M,N] = n . m
// =====================================================================
__global__ __launch_bounds__(256) void varifold_wmma_kernel(
    const float* __restrict__ xyz1, const float* __restrict__ xyz2,
    const float* __restrict__ nor1, const float* __restrict__ nor2,
    float* __restrict__ partials)
{
    const int tiles_per_batch = NPTS / 16;            // 256
    const int wave  = (blockIdx.x * blockDim.x + threadIdx.x) >> 5;
    const int lane  = threadIdx.x & 31;
    const int b     = wave / tiles_per_batch;         // wave-uniform
    const int itile = wave % tiles_per_batch;

    const int row = lane & 15;    // M within tile / N within tile (B side)
    const int hi  = lane >> 4;    // 0 -> K0,K1 ; 1 -> K2,K3

    // ---- A fragments (loaded once) ----
    const int gi = itile * 16 + row;
    const float* xb = xyz1 + ((size_t)b * NPTS + gi) * 3;
    const float x0 = xb[0], x1 = xb[1], x2 = xb[2];
    const float sqx = x0 * x0 + x1 * x1 + x2 * x2;
    const float* nb = nor1 + ((size_t)b * NPTS + gi) * 3;
    const float n0 = nb[0], n1 = nb[1], n2 = nb[2];

    v2f a_pos, a_nor;
    a_pos.x = hi ? (2.0f * x2) : (2.0f * x0);
    a_pos.y = hi ? (-sqx)      : (2.0f * x1);
    a_nor.x = hi ? n2   : n0;
    a_nor.y = hi ? 0.0f : n1;

    const float* yc = xyz2 + (size_t)b * NPTS * 3;
    const float* mc = nor2 + (size_t)b * NPTS * 3;

    float sum = 0.0f;
    for (int jt = 0; jt < tiles_per_batch; ++jt) {
        const int gj = jt * 16 + row;
        const float y0 = yc[gj * 3 + 0];
        const float y1 = yc[gj * 3 + 1];
        const float y2 = yc[gj * 3 + 2];
        const float sqy = y0 * y0 + y1 * y1 + y2 * y2; // column constant (N = lane&15)
        const float m0 = mc[gj * 3 + 0];
        const float m1 = mc[gj * 3 + 1];
        const float m2 = mc[gj * 3 + 2];

        v2f b_pos, b_nor;
        b_pos.x = hi ? y2   : y0;
        b_pos.y = hi ? 1.0f : y1;
        b_nor.x = hi ? m2   : m0;
        b_nor.y = hi ? 0.0f : m1;

        v8f c0 = {}; // 2 x.y - |x|^2
        v8f c1 = {}; // n.m
        c0 = __builtin_amdgcn_wmma_f32_16x16x4_f32(
                false, a_pos, false, b_pos, (short)0, c0, false, false);
        c1 = __builtin_amdgcn_wmma_f32_16x16x4_f32(
                false, a_nor, false, b_nor, (short)0, c1, false, false);

#pragma unroll
        for (int r = 0; r < 8; ++r) {
            const float g  = c0[r] - sqy;   // = -d2 (sigma = 1)
            const float nd = c1[r];
            sum += __expf(g) * nd * nd;
        }
    }

    // wave32 reduction
#pragma unroll
    for (int off = 16; off >= 1; off >>= 1)
        sum += __shfl_xor(sum, off, 32);

    if (lane == 0)
        partials[b * tiles_per_batch + itile] = sum;
}

// =====================================================================
// Kernel 3: deterministic reduction of per-itile partials -> out[b].
// =====================================================================
__global__ __launch_bounds__(256) void reduce_partials_kernel(
    const float* __restrict__ partials, float* __restrict__ out)
{
    __shared__ float s[256];
    const int b = blockIdx.x;
    s[threadIdx.x] = partials[b * 256 + threadIdx.x];
    __syncthreads();
#pragma unroll
    for (int off = 128; off > 0; off >>= 1) {
        if (threadIdx.x < off) s[threadIdx.x] += s[threadIdx.x + off];
        __syncthreads();
    }
    if (threadIdx.x == 0) out[b] = s[0];
}

// =====================================================================
extern "C" void kernel_launch(void* const* d_in, const int* in_sizes, int n_in,
                              void* d_out, int out_size, void* d_ws, size_t ws_size,
                              hipStream_t stream)
{
    const float* xyz1 = (const float*)d_in[0];
    const float* xyz2 = (const float*)d_in[1];
    float* out = (float*)d_out;

    // workspace layout: nor1 | nor2 | partials
    float* nor1     = (float*)d_ws;
    float* nor2     = nor1 + (size_t)BATCH * NPTS * 3;
    float* partials = nor2 + (size_t)BATCH * NPTS * 3;

    const int nblocks_normals = BATCH * (NPTS / 256);   // 64
    estimate_normals_kernel<<<nblocks_normals, 256, 0, stream>>>(xyz1, nor1);
    estimate_normals_kernel<<<nblocks_normals, 256, 0, stream>>>(xyz2, nor2);

    const int total_waves = BATCH * (NPTS / 16);        // 1024 waves
    const int nblocks_var = total_waves * 32 / 256;     // 128 blocks
    varifold_wmma_kernel<<<nblocks_var, 256, 0, stream>>>(
        xyz1, xyz2, nor1, nor2, partials);

    reduce_partials_kernel<<<BATCH, 256, 0, stream>>>(partials, out);
}